// Voxelization_44770739093796
// MI455X (gfx1250) — compile-verified
//
#include <hip/hip_runtime.h>
#include <hip/hip_bf16.h>
#include <stdint.h>

// ---------------- problem constants (from reference) ----------------
#define BATCH   8
#define NPTS    300000
#define CDIM    4
#define MAXP    5
#define MAXV    20000
#define GXD     1408
#define GYD     1600
#define GZD     40
#define GXY     (GXD * GYD)          // 2,252,800
#define VSX     0.05f
#define VSY     0.05f
#define VSZ     0.10f
#define LOX     0.0f
#define LOY     (-40.0f)
#define LOZ     (-3.0f)

#define HASH_SZ (1 << 19)            // per-batch hash, load factor <= 0.57
#define HASH_MASK (HASH_SZ - 1)
#define IMAX    0x7fffffff

#define NB512   ((NPTS + 511) / 512) // 586 scan blocks per batch
#define SCAN_SLAB 1024               // padded per-batch block-sum slab

// output layout in d_out (floats), reference tuple order
#define VOX_OFF   0
#define COORS_OFF (BATCH * MAXV * MAXP * CDIM)            // 3,200,000
#define NUM_OFF   (COORS_OFF + BATCH * MAXV * 4)          // 3,840,000
#define CEN_OFF   (NUM_OFF + BATCH * MAXV)                // 4,000,000

// ---------------- init kernels ----------------
__global__ void vox_init_out(float* __restrict__ out) {
    int r = blockIdx.x * blockDim.x + threadIdx.x;   // row over B*MAXV
    if (r >= BATCH * MAXV) return;
    int b = r / MAXV;
    float* vox = out + VOX_OFF + (size_t)r * MAXP * CDIM;
#pragma unroll
    for (int k = 0; k < MAXP * CDIM; ++k) vox[k] = 0.0f;
    float* co = out + COORS_OFF + (size_t)r * 4;
    co[0] = (float)b; co[1] = -1.0f; co[2] = -1.0f; co[3] = -1.0f;
    out[NUM_OFF + r] = 0.0f;
    float* ce = out + CEN_OFF + (size_t)r * 3;
    ce[0] = -0.5f * VSX + LOX;       // centers for coors == -1
    ce[1] = -0.5f * VSY + LOY;
    ce[2] = -0.5f * VSZ + LOZ;
}

__global__ void vox_init_ws(int* __restrict__ hashKey, int* __restrict__ hashFirst,
                            int* __restrict__ hashCount, int* __restrict__ minIdx,
                            int* __restrict__ placed, int* __restrict__ markH) {
    int wsb = blockIdx.y;
    int i = blockIdx.x * blockDim.x + threadIdx.x;
    if (i < HASH_SZ) {
        size_t e = (size_t)wsb * HASH_SZ + i;
        hashKey[e]   = -1;
        hashFirst[e] = IMAX;
        hashCount[e] = 0;
        minIdx[e]    = IMAX;
    }
    if (i < NPTS) {
        size_t e = (size_t)wsb * NPTS + i;
        placed[e] = 0;
        markH[e]  = -1;
    }
}

// ---------------- K1: coord + hash insert (async LDS staging) ----------------
__global__ void vox_assign(const float* __restrict__ pts, int b0,
                           int* __restrict__ hashKey, int* __restrict__ hashFirst,
                           int* __restrict__ hashCount, int* __restrict__ pointH) {
    __shared__ float shm[256 * 4];
    int wsb = blockIdx.y;
    int b   = b0 + wsb;
    int tid = threadIdx.x;
    int i   = blockIdx.x * 256 + tid;
    if (i >= NPTS) return;

    const float* base = pts + (size_t)b * NPTS * CDIM;  // block-uniform -> SGPR pair

    // CDNA5 async global->LDS copy: each lane moves its 16B point into LDS,
    // then the wave waits on ASYNCcnt (gfx1250 split wait counters).
    unsigned lds_addr = (unsigned)(size_t)(void*)(&shm[tid * 4]);
    unsigned voff     = (unsigned)(i * 16);
    asm volatile("global_load_async_to_lds_b128 %0, %1, %2"
                 :: "v"(lds_addr), "v"(voff),
                    "s"((unsigned long long)(uintptr_t)base)
                 : "memory");
    asm volatile("s_wait_asynccnt 0" ::: "memory");

    float px = shm[tid * 4 + 0];
    float py = shm[tid * 4 + 1];
    float pz = shm[tid * 4 + 2];

    int cx = (int)floorf((px - LOX) / VSX);
    int cy = (int)floorf((py - LOY) / VSY);
    int cz = (int)floorf((pz - LOZ) / VSZ);

    bool valid = (cx >= 0) && (cx < GXD) && (cy >= 0) && (cy < GYD) &&
                 (cz >= 0) && (cz < GZD);
    size_t pe = (size_t)wsb * NPTS + i;
    if (!valid) { pointH[pe] = -1; return; }

    int lid = cz * GXY + cy * GXD + cx;

    size_t hb = (size_t)wsb * HASH_SZ;
    unsigned h = ((unsigned)lid * 2654435761u) & HASH_MASK;
    for (;;) {
        int prev = atomicCAS(&hashKey[hb + h], -1, lid);
        if (prev == -1 || prev == lid) break;
        h = (h + 1) & HASH_MASK;
    }
    atomicMin(&hashFirst[hb + h], i);
    atomicAdd(&hashCount[hb + h], 1);
    pointH[pe] = (int)h;
}

// ---------------- K2: mark first-appearance positions ----------------
__global__ void vox_mark(const int* __restrict__ hashKey,
                         const int* __restrict__ hashFirst,
                         int* __restrict__ markH) {
    int wsb = blockIdx.y;
    int i = blockIdx.x * blockDim.x + threadIdx.x;
    if (i >= HASH_SZ) return;
    size_t e = (size_t)wsb * HASH_SZ + i;
    if (hashKey[e] != -1) markH[(size_t)wsb * NPTS + hashFirst[e]] = i; // firsts unique
}

// ---------------- K3: rank by first appearance (segmented prefix sum) ------
__global__ void vox_blocksum(const int* __restrict__ markH, int* __restrict__ blockSums) {
    __shared__ int red[512];
    int wsb = blockIdx.y;
    int tid = threadIdx.x;
    int i   = blockIdx.x * 512 + tid;
    red[tid] = (i < NPTS && markH[(size_t)wsb * NPTS + i] >= 0) ? 1 : 0;
    __syncthreads();
#pragma unroll
    for (int s = 256; s > 0; s >>= 1) {
        if (tid < s) red[tid] += red[tid + s];
        __syncthreads();
    }
    if (tid == 0) blockSums[wsb * SCAN_SLAB + blockIdx.x] = red[0];
}

__global__ void vox_scanblocks(const int* __restrict__ blockSums,
                               int* __restrict__ blockOffs, int nbatches) {
    int t = threadIdx.x;                    // one thread per batch slab
    if (t >= nbatches) return;
    int acc = 0;
    for (int j = 0; j < NB512; ++j) {
        blockOffs[t * SCAN_SLAB + j] = acc;
        acc += blockSums[t * SCAN_SLAB + j];
    }
}

__global__ void vox_slots(const float* __restrict__ pts, int b0,
                          const int* __restrict__ markH, const int* __restrict__ blockOffs,
                          const int* __restrict__ hashKey, const int* __restrict__ hashCount,
                          int* __restrict__ hashSlot, int* __restrict__ placed,
                          float* __restrict__ out) {
    __shared__ int red[512];
    int wsb = blockIdx.y;
    int b   = b0 + wsb;
    int tid = threadIdx.x;
    int i   = blockIdx.x * 512 + tid;
    int f   = (i < NPTS && markH[(size_t)wsb * NPTS + i] >= 0) ? 1 : 0;
    red[tid] = f;
    __syncthreads();
    for (int s = 1; s < 512; s <<= 1) {
        int t = (tid >= s) ? red[tid - s] : 0;
        __syncthreads();
        red[tid] += t;
        __syncthreads();
    }
    if (!f) return;
    int rank = blockOffs[wsb * SCAN_SLAB + blockIdx.x] + red[tid] - 1; // exclusive rank
    int h = markH[(size_t)wsb * NPTS + i];
    size_t he = (size_t)wsb * HASH_SZ + h;
    if (rank >= MAXV) { hashSlot[he] = -1; return; }
    hashSlot[he] = rank;

    int lid = hashKey[he];
    int cz  = lid / GXY;
    int rem = lid - cz * GXY;
    int cy  = rem / GXD;
    int cx  = rem - cy * GXD;

    int row = b * MAXV + rank;
    float* co = out + COORS_OFF + (size_t)row * 4;
    co[0] = (float)b; co[1] = (float)cz; co[2] = (float)cy; co[3] = (float)cx;
    int cnt = hashCount[he];
    out[NUM_OFF + row] = (float)(cnt < MAXP ? cnt : MAXP);
    float* ce = out + CEN_OFF + (size_t)row * 3;
    ce[0] = ((float)cx + 0.5f) * VSX + LOX;
    ce[1] = ((float)cy + 0.5f) * VSY + LOY;
    ce[2] = ((float)cz + 0.5f) * VSZ + LOZ;

    // i IS this voxel's first-appearance point: place it at within==0 now
    // (with uniform points most voxels have exactly one point, so this
    //  removes nearly all tournament work).
    const float4 v = *(const float4*)(pts + ((size_t)b * NPTS + i) * CDIM);
    float* dst = out + VOX_OFF + ((size_t)row * MAXP + 0) * CDIM;
    *(float4*)dst = v;
    placed[(size_t)wsb * NPTS + i] = 1;
}

// ---------------- K4: deterministic next-smallest selection per voxel ------
__global__ void vox_minpass(const int* __restrict__ pointH, const int* __restrict__ placed,
                            const int* __restrict__ hashSlot, int* __restrict__ minIdx) {
    int wsb = blockIdx.y;
    int i = blockIdx.x * blockDim.x + threadIdx.x;
    if (i >= NPTS) return;
    size_t pe = (size_t)wsb * NPTS + i;
    int h = pointH[pe];
    if (h < 0 || placed[pe]) return;
    size_t he = (size_t)wsb * HASH_SZ + h;
    if (hashSlot[he] < 0) return;
    atomicMin(&minIdx[he], i);
}

__global__ void vox_placepass(const float* __restrict__ pts, int b0, int p,
                              const int* __restrict__ pointH, int* __restrict__ placed,
                              const int* __restrict__ hashSlot, int* __restrict__ minIdx,
                              float* __restrict__ out) {
    int wsb = blockIdx.y;
    int b   = b0 + wsb;
    int i = blockIdx.x * blockDim.x + threadIdx.x;
    if (i >= NPTS) return;
    size_t pe = (size_t)wsb * NPTS + i;
    int h = pointH[pe];
    if (h < 0 || placed[pe]) return;
    size_t he = (size_t)wsb * HASH_SZ + h;
    int slot = hashSlot[he];
    if (slot < 0) return;
    if (minIdx[he] != i) return;
    // this lane holds the p-th smallest original index in this voxel
    placed[pe] = 1;
    minIdx[he] = IMAX;   // reset for next pass
    const float4 v = *(const float4*)(pts + ((size_t)b * NPTS + i) * CDIM);
    float* dst = out + VOX_OFF + (((size_t)(b * MAXV + slot)) * MAXP + p) * CDIM;
    *(float4*)dst = v;
}

// ---------------- launcher ----------------
extern "C" void kernel_launch(void* const* d_in, const int* in_sizes, int n_in,
                              void* d_out, int out_size, void* d_ws, size_t ws_size,
                              hipStream_t stream) {
    const float* pts = (const float*)d_in[0];
    float* out = (float*)d_out;

    // workspace need per batch-slab
    const size_t perHash = (size_t)HASH_SZ * 4;
    const size_t perPts  = (size_t)NPTS * 4;
    const size_t perScan = (size_t)SCAN_SLAB * 4;
    const size_t slabBytes = 5 * perHash + 3 * perPts + 2 * perScan;

    // deterministic host-side choice: 8 concurrent slabs if workspace allows
    const int NSLAB = (ws_size >= (size_t)BATCH * slabBytes) ? BATCH : 1;

    char* ws = (char*)d_ws;
    int* hashKey   = (int*)ws; ws += (size_t)NSLAB * perHash;
    int* hashFirst = (int*)ws; ws += (size_t)NSLAB * perHash;
    int* hashCount = (int*)ws; ws += (size_t)NSLAB * perHash;
    int* hashSlot  = (int*)ws; ws += (size_t)NSLAB * perHash;
    int* minIdx    = (int*)ws; ws += (size_t)NSLAB * perHash;
    int* pointH    = (int*)ws; ws += (size_t)NSLAB * perPts;
    int* placed    = (int*)ws; ws += (size_t)NSLAB * perPts;
    int* markH     = (int*)ws; ws += (size_t)NSLAB * perPts;
    int* blockSums = (int*)ws; ws += (size_t)NSLAB * perScan;
    int* blockOffs = (int*)ws; ws += (size_t)NSLAB * perScan;

    const int NB256 = (NPTS + 255) / 256;     // 1172
    const int NBH   = (HASH_SZ + 255) / 256;  // 2048
    const int NBROW = (BATCH * MAXV + 255) / 256;

    vox_init_out<<<NBROW, 256, 0, stream>>>(out);

    for (int b0 = 0; b0 < BATCH; b0 += NSLAB) {
        dim3 gH(NBH, NSLAB), gP(NB256, NSLAB), gS(NB512, NSLAB);
        vox_init_ws<<<gH, 256, 0, stream>>>(hashKey, hashFirst, hashCount,
                                            minIdx, placed, markH);
        vox_assign<<<gP, 256, 0, stream>>>(pts, b0, hashKey, hashFirst,
                                           hashCount, pointH);
        vox_mark<<<gH, 256, 0, stream>>>(hashKey, hashFirst, markH);
        vox_blocksum<<<gS, 512, 0, stream>>>(markH, blockSums);
        vox_scanblocks<<<1, 32, 0, stream>>>(blockSums, blockOffs, NSLAB);
        vox_slots<<<gS, 512, 0, stream>>>(pts, b0, markH, blockOffs, hashKey,
                                          hashCount, hashSlot, placed, out);
        for (int p = 1; p < MAXP; ++p) {   // within==0 already placed in vox_slots
            vox_minpass<<<gP, 256, 0, stream>>>(pointH, placed, hashSlot, minIdx);
            vox_placepass<<<gP, 256, 0, stream>>>(pts, b0, p, pointH, placed,
                                                  hashSlot, minIdx, out);
        }
    }
}